// Prox_20667382628975
// MI455X (gfx1250) — compile-verified
//
#include <hip/hip_runtime.h>
#include <hip/hip_bf16.h>
#include <stdint.h>

// Problem shape (from reference): x is (16, 8, 512, 512) f32
#define NB 16
#define NC 8
#define NH 512
#define NW 512
#define NROWS (NB * NC)      // 128 independent rows
#define ROWN  (NH * NW)      // 262144 elements per row

// 0-indexed order statistics needed for q=0.01 and q=0.99 with linear interp:
// pos = q*(n-1):  0.01*262143 = 2621.43 ; 0.99*262143 = 259521.57
#define R1A 2621u
#define R1B 2622u
#define R9A 259521u
#define R9B 259522u

typedef float f4 __attribute__((ext_vector_type(4)));

__device__ __forceinline__ uint32_t f2key(float f) {
    uint32_t u = __float_as_uint(f);
    return (u & 0x80000000u) ? ~u : (u | 0x80000000u);
}
__device__ __forceinline__ float key2f(uint32_t k) {
    uint32_t u = (k & 0x80000000u) ? (k & 0x7fffffffu) : ~k;
    return __uint_as_float(u);
}

// ---------------------------------------------------------------------------
// Pass 1: per-row 2^16-bin histogram of the top 16 key bits in 256KB LDS
// (CDNA5-only: 320KB LDS per WGP). One workgroup per WGP is forced by the LDS
// footprint, so run the max 1024 threads (32 waves, 8/SIMD) to hide HBM +
// LDS-atomic latency. A Hillis-Steele block scan then locates the bin and
// count-before for all 4 target ranks in parallel.
// ---------------------------------------------------------------------------
__global__ void __launch_bounds__(1024)
hist_top16(const float* __restrict__ x,
           uint32_t* __restrict__ pref,   // [NROWS*4] key prefix (16 bits)
           uint32_t* __restrict__ cbef)   // [NROWS*4] count below prefix bin
{
    extern __shared__ uint32_t sh[];       // 65536 bins + 1024 scan slots
    uint32_t* hist = sh;
    uint32_t* part = sh + 65536;
    const int row = blockIdx.x;
    const int tid = threadIdx.x;           // 1024 threads

    for (int i = tid; i < 65536; i += 1024) hist[i] = 0u;
    __syncthreads();

    const f4* p = (const f4*)(x + (size_t)row * ROWN);
    for (int i = 0; i < ROWN / 4; i += 1024) {
        // gfx1250 prefetch ahead (~64KB) into L2 for the streaming read
        __builtin_prefetch((const void*)(p + i + tid + 4096), 0, 1);
        f4 v = p[i + tid];
        atomicAdd(&hist[f2key(v.x) >> 16], 1u);
        atomicAdd(&hist[f2key(v.y) >> 16], 1u);
        atomicAdd(&hist[f2key(v.z) >> 16], 1u);
        atomicAdd(&hist[f2key(v.w) >> 16], 1u);
    }
    __syncthreads();

    // per-thread partial over 64 bins, then block-wide inclusive scan
    const int base = tid * 64;
    uint32_t s = 0;
    for (int j = 0; j < 64; ++j) s += hist[base + j];
    part[tid] = s;
    __syncthreads();
    for (int off = 1; off < 1024; off <<= 1) {
        uint32_t v = (tid >= off) ? part[tid - off] : 0u;
        __syncthreads();
        part[tid] += v;
        __syncthreads();
    }
    uint32_t cum = part[tid] - s;          // exclusive prefix for this chunk

    const uint32_t ranks[4] = {R1A, R1B, R9A, R9B};
    for (int j = 0; j < 64; ++j) {
        uint32_t c = hist[base + j];
        if (c) {
            #pragma unroll
            for (int k = 0; k < 4; ++k)
                if (ranks[k] >= cum && ranks[k] < cum + c) {
                    pref[row * 4 + k] = (uint32_t)(base + j);
                    cbef[row * 4 + k] = cum;
                }
        }
        cum += c;
    }
}

// ---------------------------------------------------------------------------
// Passes 2 & 3: refine 8 more key bits per pass. x is L2-resident after pass 1
// (128MB < 192MB L2), so these re-reads never touch HBM. Pass 3 (final_)
// reconstructs the exact order statistics and the per-row gate parameters.
// ---------------------------------------------------------------------------
__global__ void __launch_bounds__(512)
refine8(const float* __restrict__ x,
        uint32_t* __restrict__ pref,
        uint32_t* __restrict__ cbef,
        const float* __restrict__ alpha,
        const float* __restrict__ beta,
        float* __restrict__ thm,   // [NROWS] th * mask
        float* __restrict__ scl,   // [NROWS] beta / th_new
        int pshift, int final_)
{
    __shared__ uint32_t h[4 * 256];
    __shared__ uint32_t sp[4], sc[4];
    __shared__ float    sv[4];
    const int row = blockIdx.x;
    const int tid = threadIdx.x;           // 512 threads

    if (tid < 4) { sp[tid] = pref[row * 4 + tid]; sc[tid] = cbef[row * 4 + tid]; }
    for (int i = tid; i < 1024; i += 512) h[i] = 0u;
    __syncthreads();

    const uint32_t p0 = sp[0], p1 = sp[1], p2 = sp[2], p3 = sp[3];
    const int bsh = pshift - 8;
    const f4* px = (const f4*)(x + (size_t)row * ROWN);

    for (int i = tid; i < ROWN / 4; i += 512) {
        f4 v = px[i];
        #pragma unroll
        for (int c = 0; c < 4; ++c) {
            uint32_t k  = f2key(v[c]);
            uint32_t kp = k >> pshift;
            uint32_t kb = (k >> bsh) & 255u;
            if (kp == p0) atomicAdd(&h[0 * 256 + kb], 1u);
            if (kp == p1) atomicAdd(&h[1 * 256 + kb], 1u);
            if (kp == p2) atomicAdd(&h[2 * 256 + kb], 1u);
            if (kp == p3) atomicAdd(&h[3 * 256 + kb], 1u);
        }
    }
    __syncthreads();

    if (tid < 4) {
        const uint32_t ranks[4] = {R1A, R1B, R9A, R9B};
        uint32_t r   = ranks[tid];
        uint32_t cum = sc[tid];
        uint32_t bin = 255u;
        for (int j = 0; j < 256; ++j) {
            uint32_t c = h[tid * 256 + j];
            if (r < cum + c) { bin = (uint32_t)j; break; }
            cum += c;
        }
        uint32_t np = (sp[tid] << 8) | bin;
        if (final_) sv[tid] = key2f(np);
        else { pref[row * 4 + tid] = np; cbef[row * 4 + tid] = cum; }
    }
    __syncthreads();

    if (final_ && tid == 0) {
        double pos1 = 0.01 * (double)(ROWN - 1);
        double pos9 = 0.99 * (double)(ROWN - 1);
        float f1 = (float)(pos1 - floor(pos1));
        float f9 = (float)(pos9 - floor(pos9));
        float i1  = sv[0] + f1 * (sv[1] - sv[0]);
        float i99 = sv[2] + f9 * (sv[3] - sv[2]);
        float th  = i1 + (i99 - i1) * alpha[0];
        float m   = (th > 1e-14f) ? 1.0f : 0.0f;
        float thn = th * m + (1.0f - m);
        thm[row] = th * m;
        scl[row] = beta[0] / thn;
    }
}

// ---------------------------------------------------------------------------
// Pass 4: gated ReLU. Double-buffered gfx1250 async global->LDS staging
// (ASYNCcnt pipeline, no VGPR return traffic), v_exp_f32/v_rcp_f32 sigmoid,
// non-temporal B128 stores so the 128MB output stream does not evict x's
// L2-resident lines while we read them.
// ---------------------------------------------------------------------------
__global__ void __launch_bounds__(256)
gate_kernel(const float* __restrict__ x,
            const float* __restrict__ thm,
            const float* __restrict__ scl,
            float* __restrict__ out)
{
    extern __shared__ char stage[];        // 2 x 4096B double buffer
    const int row = blockIdx.y;
    const int tid = threadIdx.x;           // 256 threads
    const float T = thm[row];
    const float S = scl[row];
    const float* xr   = x   + (size_t)row * ROWN;
    float*       orow = out + (size_t)row * ROWN;

    const int TILES = 16;                  // 16 tiles * 1024 floats = 16384 floats/block
    const uint32_t laneByte  = (uint32_t)tid * 16u;
    const uint32_t blockByte = (uint32_t)blockIdx.x * (uint32_t)(TILES * 4096);

    // prologue: stage tile 0
    asm volatile("global_load_async_to_lds_b128 %0, %1, %2"
                 :: "v"(laneByte), "v"(blockByte + laneByte), "s"(xr) : "memory");

    for (int t = 0; t < TILES; ++t) {
        if (t + 1 < TILES) {
            // buffer (t+1)&1 was consumed in iteration t-1; ensure its DS reads drained
            asm volatile("s_wait_dscnt 0x0" ::: "memory");
            asm volatile("global_load_async_to_lds_b128 %0, %1, %2"
                         :: "v"((uint32_t)((t + 1) & 1) * 4096u + laneByte),
                            "v"(blockByte + (uint32_t)(t + 1) * 4096u + laneByte),
                            "s"(xr) : "memory");
            asm volatile("s_wait_asynccnt 0x1" ::: "memory");  // tile t complete (in-order)
        } else {
            asm volatile("s_wait_asynccnt 0x0" ::: "memory");
        }

        f4 v = *(const f4*)(stage + (size_t)((t & 1) * 4096) + (size_t)tid * 16);
        f4 o;
        #pragma unroll
        for (int c = 0; c < 4; ++c) {
            float xv = v[c];
            float z  = S * (fabsf(xv) - T);
            float e  = __builtin_amdgcn_exp2f(-z * 1.44269504088896340736f); // exp(-z)
            float g  = __builtin_amdgcn_rcpf(1.0f + e);                      // sigmoid
            o[c] = fmaxf(xv, 0.0f) * g;
        }
        __builtin_nontemporal_store(
            o, (f4*)(orow + (size_t)blockIdx.x * (TILES * 1024) + (size_t)t * 1024 + (size_t)tid * 4));
    }
}

// ---------------------------------------------------------------------------
extern "C" void kernel_launch(void* const* d_in, const int* in_sizes, int n_in,
                              void* d_out, int out_size, void* d_ws, size_t ws_size,
                              hipStream_t stream) {
    const float* x     = (const float*)d_in[0];
    const float* alpha = (const float*)d_in[1];
    const float* beta  = (const float*)d_in[2];
    float*       out   = (float*)d_out;

    // workspace layout
    uint32_t* pref = (uint32_t*)d_ws;            // [NROWS*4]
    uint32_t* cbef = pref + NROWS * 4;           // [NROWS*4]
    float*    thm  = (float*)(cbef + NROWS * 4); // [NROWS]
    float*    scl  = thm + NROWS;                // [NROWS]

    const int histLds = (65536 + 1024) * (int)sizeof(uint32_t);  // 266KB < 320KB/WGP
    hipFuncSetAttribute((const void*)hist_top16,
                        hipFuncAttributeMaxDynamicSharedMemorySize, histLds);

    hist_top16<<<NROWS, 1024, histLds, stream>>>(x, pref, cbef);
    refine8<<<NROWS, 512, 0, stream>>>(x, pref, cbef, alpha, beta, thm, scl, 16, 0);
    refine8<<<NROWS, 512, 0, stream>>>(x, pref, cbef, alpha, beta, thm, scl, 8, 1);

    dim3 grid(ROWN / (16 * 1024), NROWS);   // (16, 128)
    gate_kernel<<<grid, 256, 2 * 4096, stream>>>(x, thm, scl, out);
}